// EMATranVectorQuantizer_65292092834256
// MI455X (gfx1250) — compile-verified
//
#include <hip/hip_runtime.h>

typedef float v2f __attribute__((ext_vector_type(2)));
typedef float v4f __attribute__((ext_vector_type(4)));
typedef float v8f __attribute__((ext_vector_type(8)));

constexpr int N_TOK = 65536;
constexpr int K_CB  = 128;
constexpr int D_EMB = 32;
constexpr int CB_STRIDE = 34;   // 32 + 2 pad: spreads LDS banks for the B-chunk reads

// ---------------------------------------------------------------------------
// Kernel 1: fused  dist = ||c||^2 - 2 x.c  ->  argmin_k  ->  gather + STE out
// One wave = 16 rows. GEMM via chained V_WMMA_F32_16X16X4_F32 (fp32 matrix op,
// matches reference precision). 8 code tiles x 8 K-chunks = 64 WMMA per wave.
// ---------------------------------------------------------------------------
__global__ __launch_bounds__(256)
void vq_argmin_gather_kernel(const float* __restrict__ latent,
                             const float* __restrict__ codebook,
                             float* __restrict__ outPolicy,
                             float* __restrict__ outQuant)
{
    __shared__ float sCB[K_CB * CB_STRIDE];
    __shared__ float sCN[K_CB];

    const int tid = threadIdx.x;

    // Stage codebook (16 KB) into LDS, row-major with padded stride.
    const float2* cb2 = reinterpret_cast<const float2*>(codebook);
#pragma unroll
    for (int it = 0; it < 8; ++it) {
        int idx = tid + it * 256;            // 0..2047 float2's
        int k = idx >> 4, j = idx & 15;
        *reinterpret_cast<float2*>(&sCB[k * CB_STRIDE + 2 * j]) = cb2[idx];
    }
    __syncthreads();

    // Per-code squared norms.
    if (tid < K_CB) {
        float s = 0.f;
#pragma unroll
        for (int d = 0; d < D_EMB; ++d) {
            float c = sCB[tid * CB_STRIDE + d];
            s = fmaf(c, c, s);
        }
        sCN[tid] = s;
    }
    __syncthreads();

    const int wave = tid >> 5;
    const int lane = tid & 31;
    const int m    = lane & 15;    // A row / B column within tile (ISA layout)
    const int half = lane >> 4;    // K sub-pair selector (ISA layout)
    const int base = (blockIdx.x * 8 + wave) * 16;

    // A-matrix chunks: lane holds row (base+m), K elements {kc*4+half*2, +1}.
    v2f a[8];
#pragma unroll
    for (int kc = 0; kc < 8; ++kc) {
        float2 t = *reinterpret_cast<const float2*>(
            &latent[(size_t)(base + m) * D_EMB + kc * 4 + half * 2]);
        a[kc].x = t.x; a[kc].y = t.y;
    }

    float bestV[8];
    int   bestI[8];
#pragma unroll
    for (int v = 0; v < 8; ++v) { bestV[v] = 3.0e38f; bestI[v] = 0x7fffffff; }

    // 8 tiles of 16 codes; accumulate 16x16x4 f32 WMMA over K=32.
#pragma unroll
    for (int ct = 0; ct < 8; ++ct) {
        const int code = ct * 16 + m;
        v8f acc = {};
#pragma unroll
        for (int kc = 0; kc < 8; ++kc) {
            float2 t = *reinterpret_cast<const float2*>(
                &sCB[code * CB_STRIDE + kc * 4 + half * 2]);
            v2f bv; bv.x = t.x; bv.y = t.y;
            acc = __builtin_amdgcn_wmma_f32_16x16x4_f32(
                false, a[kc], false, bv, (short)0, acc, false, false);
        }
        const float cn = sCN[code];
        // C layout: lane holds column (ct*16+m), rows v + 8*half across 8 VGPRs.
#pragma unroll
        for (int v = 0; v < 8; ++v) {
            float val = fmaf(-2.f, acc[v], cn);
            if (val < bestV[v] || (val == bestV[v] && code < bestI[v])) {
                bestV[v] = val; bestI[v] = code;
            }
        }
    }

    // Butterfly (val,idx) min-reduce within each 16-lane half (columns 0..15).
#pragma unroll
    for (int s = 0; s < 4; ++s) {
        const int msk = 1 << s;
#pragma unroll
        for (int v = 0; v < 8; ++v) {
            float ov = __shfl_xor(bestV[v], msk, 32);
            int   oi = __shfl_xor(bestI[v], msk, 32);
            if (ov < bestV[v] || (ov == bestV[v] && oi < bestI[v])) {
                bestV[v] = ov; bestI[v] = oi;
            }
        }
    }

    // Broadcast winning index per row, gather code, write STE outputs.
    // Rows 0..7 live in lanes 0-15 (take lane 0), rows 8..15 in lanes 16-31.
#pragma unroll
    for (int r = 0; r < 16; ++r) {
        const int v   = r & 7;
        const int src = (r < 8) ? 0 : 16;
        const int idx = __shfl(bestI[v], src, 32);
        const size_t row = (size_t)base + r;
        const float q = sCB[idx * CB_STRIDE + lane];     // lane == d (D==32)
        const float l = latent[row * D_EMB + lane];
        const float diff = q - l;                        // match reference op order
        outQuant [row * D_EMB + lane] = q;
        outPolicy[row * D_EMB + lane] = l + diff;
    }
}

// ---------------------------------------------------------------------------
// Kernel 2: codebook_set broadcast — 1.07 GB streaming store (the bottleneck).
// float4 non-temporal stores; source value is loop-invariant per thread.
// ---------------------------------------------------------------------------
__global__ __launch_bounds__(256)
void vq_broadcast_kernel(const v4f* __restrict__ cb,   // 1024 float4 = 16 KB
                         v4f* __restrict__ out,
                         long long total4)
{
    long long i = (long long)blockIdx.x * blockDim.x + threadIdx.x;
    const long long stride = (long long)gridDim.x * blockDim.x;
    for (; i < total4; i += stride) {
        v4f val = cb[i & 1023];
        __builtin_nontemporal_store(val, &out[i]);
    }
}

extern "C" void kernel_launch(void* const* d_in, const int* in_sizes, int n_in,
                              void* d_out, int out_size, void* d_ws, size_t ws_size,
                              hipStream_t stream)
{
    (void)in_sizes; (void)n_in; (void)out_size; (void)d_ws; (void)ws_size;

    const float* latent   = (const float*)d_in[0];   // [N, 32] f32
    const float* codebook = (const float*)d_in[1];   // [128, 32] f32

    float* out       = (float*)d_out;
    float* outPolicy = out;                                   // [N,32]
    float* outQuant  = out + (size_t)N_TOK * D_EMB;           // [N,32]
    float* outSet    = out + 2ull * N_TOK * D_EMB;            // [N,128,32]

    // 512 blocks x 8 waves x 16 rows = 65536 rows
    vq_argmin_gather_kernel<<<dim3(N_TOK / 128), dim3(256), 0, stream>>>(
        latent, codebook, outPolicy, outQuant);

    const long long total4 = (long long)N_TOK * (K_CB * D_EMB / 4);  // 67,108,864
    vq_broadcast_kernel<<<dim3(8192), dim3(256), 0, stream>>>(
        (const v4f*)codebook, (v4f*)outSet, total4);
}